// monet_polar_segmentation_31782757990673
// MI455X (gfx1250) — compile-verified
//
#include <hip/hip_runtime.h>
#include <hip/hip_bf16.h>
#include <cstddef>

typedef float v2f __attribute__((ext_vector_type(2)));
typedef float v8f __attribute__((ext_vector_type(8)));

// ---------------------------------------------------------------------------
// Dense GEMM with CDNA5 fp32 WMMA: C[M,N] = A[M,K] @ B[K,N]
// One wave computes a 32x32 output tile = 2x2 grid of 16x16 WMMA tiles.
// K stepped by 4 (all cin here are multiples of 4).
// Interior tiles take a mask-free fast path (pure loads + v_wmma); only the
// boundary fringe pays clamp+mask VALU work.
//
// Fragment layouts (CDNA5 ISA):
//   A 16x4 f32 (2 VGPR): lane<16 -> K={0,1}, lane>=16 -> K={2,3}, M=lane&15
//   B 4x16 f32 (2 VGPR): lane<16 -> K={0,1}, lane>=16 -> K={2,3}, N=lane&15
//   C/D 16x16 f32 (8 VGPR): vgpr r -> M = r + 8*(lane>=16), N = lane&15
// ---------------------------------------------------------------------------
__global__ void wmma_gemm_f32(const float* __restrict__ A, const float* __restrict__ B,
                              float* __restrict__ C, int M, int K, int N) {
  const int wavesPerBlock = blockDim.x >> 5;
  const int wave   = blockIdx.x * wavesPerBlock + ((int)threadIdx.x >> 5);
  const int ntile  = (N + 31) >> 5;               // 32-wide column tiles
  const int mtile  = (M + 31) >> 5;               // 32-tall row tiles
  if (wave >= ntile * mtile) return;              // wave-uniform exit
  const int row0 = (wave / ntile) << 5;
  const int col0 = (wave % ntile) << 5;
  const int lane = (int)threadIdx.x & 31;
  const int mn   = lane & 15;
  const int half = lane >> 4;                     // 0: K {0,1}, 1: K {2,3}

  v8f acc00 = {}, acc01 = {}, acc10 = {}, acc11 = {};

  if (row0 + 32 <= M && col0 + 32 <= N) {
    // ---------------- fast path: fully interior tile, no masks ----------------
    const float* __restrict__ arow0 = A + (size_t)(row0 + mn) * K;
    const float* __restrict__ arow1 = A + (size_t)(row0 + 16 + mn) * K;
    const int c0 = col0 + mn;
    const int c1 = col0 + 16 + mn;
    for (int k0 = 0; k0 < K; k0 += 4) {
      const int ka = k0 + half * 2;
      v2f a0, a1, b0, b1;
      a0.x = arow0[ka];
      a0.y = arow0[ka + 1];
      a1.x = arow1[ka];
      a1.y = arow1[ka + 1];
      const float* __restrict__ bk0 = B + (size_t)ka * N;
      const float* __restrict__ bk1 = B + (size_t)(ka + 1) * N;
      b0.x = bk0[c0];
      b0.y = bk1[c0];
      b1.x = bk0[c1];
      b1.y = bk1[c1];
      acc00 = __builtin_amdgcn_wmma_f32_16x16x4_f32(false, a0, false, b0, (short)0, acc00, false, false);
      acc01 = __builtin_amdgcn_wmma_f32_16x16x4_f32(false, a0, false, b1, (short)0, acc01, false, false);
      acc10 = __builtin_amdgcn_wmma_f32_16x16x4_f32(false, a1, false, b0, (short)0, acc10, false, false);
      acc11 = __builtin_amdgcn_wmma_f32_16x16x4_f32(false, a1, false, b1, (short)0, acc11, false, false);
    }
#pragma unroll
    for (int r = 0; r < 8; ++r) {
      const int orow0 = row0 + r + half * 8;
      const int orow1 = row0 + 16 + r + half * 8;
      C[(size_t)orow0 * N + c0] = acc00[r];
      C[(size_t)orow0 * N + c1] = acc01[r];
      C[(size_t)orow1 * N + c0] = acc10[r];
      C[(size_t)orow1 * N + c1] = acc11[r];
    }
  } else {
    // -------- boundary tile: clamp addresses in-range, mask values to 0 --------
    int r0 = row0 + mn;
    int r1 = row0 + 16 + mn;
    const float am0 = (r0 < M) ? 1.0f : 0.0f;
    const float am1 = (r1 < M) ? 1.0f : 0.0f;
    r0 = (r0 < M) ? r0 : (M - 1);
    r1 = (r1 < M) ? r1 : (M - 1);
    const float* __restrict__ arow0 = A + (size_t)r0 * K;
    const float* __restrict__ arow1 = A + (size_t)r1 * K;

    int c0 = col0 + mn;
    int c1 = col0 + 16 + mn;
    const float bm0 = (c0 < N) ? 1.0f : 0.0f;
    const float bm1 = (c1 < N) ? 1.0f : 0.0f;
    c0 = (c0 < N) ? c0 : (N - 1);
    c1 = (c1 < N) ? c1 : (N - 1);

    for (int k0 = 0; k0 < K; k0 += 4) {
      const int ka = k0 + half * 2;
      v2f a0, a1, b0, b1;
      a0.x = arow0[ka] * am0;
      a0.y = arow0[ka + 1] * am0;
      a1.x = arow1[ka] * am1;
      a1.y = arow1[ka + 1] * am1;
      const float* __restrict__ bk0 = B + (size_t)ka * N;
      const float* __restrict__ bk1 = B + (size_t)(ka + 1) * N;
      b0.x = bk0[c0] * bm0;
      b0.y = bk1[c0] * bm0;
      b1.x = bk0[c1] * bm1;
      b1.y = bk1[c1] * bm1;
      acc00 = __builtin_amdgcn_wmma_f32_16x16x4_f32(false, a0, false, b0, (short)0, acc00, false, false);
      acc01 = __builtin_amdgcn_wmma_f32_16x16x4_f32(false, a0, false, b1, (short)0, acc01, false, false);
      acc10 = __builtin_amdgcn_wmma_f32_16x16x4_f32(false, a1, false, b0, (short)0, acc10, false, false);
      acc11 = __builtin_amdgcn_wmma_f32_16x16x4_f32(false, a1, false, b1, (short)0, acc11, false, false);
    }

    const int oc0 = col0 + mn;
    const int oc1 = col0 + 16 + mn;
#pragma unroll
    for (int r = 0; r < 8; ++r) {
      const int orow0 = row0 + r + half * 8;
      const int orow1 = row0 + 16 + r + half * 8;
      if (orow0 < M) {
        if (oc0 < N) C[(size_t)orow0 * N + oc0] = acc00[r];
        if (oc1 < N) C[(size_t)orow0 * N + oc1] = acc01[r];
      }
      if (orow1 < M) {
        if (oc0 < N) C[(size_t)orow1 * N + oc0] = acc10[r];
        if (oc1 < N) C[(size_t)orow1 * N + oc1] = acc11[r];
      }
    }
  }
}

// ---------------------------------------------------------------------------
// Per-edge GMM message + scatter-add. One wave per edge (wave32).
// Z[n, k*cout + c] = (x @ g) precomputed.  w_k = exp(-0.5*sum_d (p-mu)^2/(s^2+eps))
// ---------------------------------------------------------------------------
__global__ void gmm_edge_kernel(const float* __restrict__ Z, const float* __restrict__ pseudo,
                                const int* __restrict__ ei, int E, int cout,
                                const float* __restrict__ mu, const float* __restrict__ sigma,
                                float* __restrict__ agg, float* __restrict__ cnt) {
  const int wavesPerBlock = blockDim.x >> 5;
  const int e = blockIdx.x * wavesPerBlock + ((int)threadIdx.x >> 5);
  if (e >= E) return;
  const int lane = (int)threadIdx.x & 31;
  const int src = ei[e];
  const int dst = ei[E + e];
  const float p0 = pseudo[2 * e];
  const float p1 = pseudo[2 * e + 1];
  float w[3];
#pragma unroll
  for (int k = 0; k < 3; ++k) {
    const float d0 = p0 - mu[2 * k];
    const float d1 = p1 - mu[2 * k + 1];
    const float s0 = sigma[2 * k];
    const float s1 = sigma[2 * k + 1];
    w[k] = expf(-0.5f * (d0 * d0 / (1e-15f + s0 * s0) + d1 * d1 / (1e-15f + s1 * s1)));
  }
  const float* z = Z + (size_t)src * (3 * cout);
  float* a = agg + (size_t)dst * cout;
  for (int c = lane; c < cout; c += 32) {
    const float msg = w[0] * z[c] + w[1] * z[cout + c] + w[2] * z[2 * cout + c];
    atomicAdd(&a[c], msg);
  }
  if (lane == 0) atomicAdd(&cnt[dst], 1.0f);
}

// out = [relu]( agg / max(cnt,1) + zr + bias )
__global__ void combine_kernel(const float* __restrict__ agg, const float* __restrict__ cnt,
                               const float* __restrict__ zr, const float* __restrict__ bias,
                               float* __restrict__ out, int n, int cout, int doRelu) {
  const int i = blockIdx.x * blockDim.x + (int)threadIdx.x;
  if (i >= n * cout) return;
  const int row = i / cout;
  const int c = i - row * cout;
  float v = agg[i] / fmaxf(cnt[row], 1.0f) + zr[i] + bias[c];
  if (doRelu) v = fmaxf(v, 0.0f);
  out[i] = v;
}

// hex max-pool with first-occurrence argmax (matches jnp.argmax), keep first L rows
__global__ void hex_pool_kernel(const float* __restrict__ x, const int* __restrict__ hexes,
                                int C, int L, float* __restrict__ vals, int* __restrict__ idx) {
  const int i = blockIdx.x * blockDim.x + (int)threadIdx.x;
  if (i >= L * C) return;
  const int n = i / C;
  const int c = i - n * C;
  const int* h = hexes + (size_t)n * 7;
  float best = x[(size_t)h[0] * C + c];
  int bj = 0;
#pragma unroll
  for (int j = 1; j < 7; ++j) {
    const float v = x[(size_t)h[j] * C + c];
    if (v > best) { best = v; bj = j; }
  }
  vals[i] = best;
  idx[i] = h[bj];
}

// cin_buf[:,0:C] = 0 (unpool target), cin_buf[:,C:Ctot] = skip  (fused zero+concat)
__global__ void unpool_prep_kernel(const float* __restrict__ skip, float* __restrict__ cin_buf,
                                   int L2, int C, int Ctot) {
  const int i = blockIdx.x * blockDim.x + (int)threadIdx.x;
  if (i >= L2 * Ctot) return;
  const int n = i / Ctot;
  const int c = i - n * Ctot;
  cin_buf[i] = (c < C) ? 0.0f : skip[(size_t)n * (Ctot - C) + (c - C)];
}

// cin_buf[idx[n,c], c] = xsmall[n,c]
__global__ void unpool_scatter_kernel(const float* __restrict__ xsmall, const int* __restrict__ idx,
                                      float* __restrict__ cin_buf, int L, int C, int Ctot) {
  const int i = blockIdx.x * blockDim.x + (int)threadIdx.x;
  if (i >= L * C) return;
  const int c = i % C;
  cin_buf[(size_t)idx[i] * Ctot + c] = xsmall[i];
}

__global__ void softmax21_kernel(const float* __restrict__ in, float* __restrict__ out, int n) {
  const int row = blockIdx.x * blockDim.x + (int)threadIdx.x;
  if (row >= n) return;
  const float* r = in + (size_t)row * 21;
  float m = r[0];
#pragma unroll
  for (int c = 1; c < 21; ++c) m = fmaxf(m, r[c]);
  float e[21];
  float s = 0.0f;
#pragma unroll
  for (int c = 0; c < 21; ++c) { e[c] = expf(r[c] - m); s += e[c]; }
  const float inv = 1.0f / s;
#pragma unroll
  for (int c = 0; c < 21; ++c) out[(size_t)row * 21 + c] = e[c] * inv;
}

// ---------------------------------------------------------------------------
// Host-side driver
// ---------------------------------------------------------------------------
namespace {
constexpr int N6 = 40962, N5 = 10242, N4 = 2562, N3 = 642, N2 = 162;
constexpr int E6 = 245760, E5 = 61440, E4 = 15360, E3 = 3840, E2 = 960;

static inline void launch_gemm(const float* A, const float* B, float* C,
                               int M, int K, int N, hipStream_t stream) {
  const int tiles = ((M + 31) / 32) * ((N + 31) / 32);
  const int blocks = (tiles + 3) / 4;                 // 4 waves / 128 threads per block
  wmma_gemm_f32<<<blocks, 128, 0, stream>>>(A, B, C, M, K, N);
}

static void run_gmm_conv(const float* x, int n, int cin, int cout,
                         const int* ei, int E, const float* pseudo,
                         const float* g, const float* mu, const float* sigma,
                         const float* root, const float* bias,
                         float* out, int doRelu,
                         float* Z, float* ZR, float* AGG, float* CNT,
                         hipStream_t stream) {
  launch_gemm(x, g, Z, n, cin, 3 * cout, stream);     // Z  = x @ g      [n, 3*cout]
  launch_gemm(x, root, ZR, n, cin, cout, stream);     // ZR = x @ root   [n, cout]
  hipMemsetAsync(AGG, 0, (size_t)n * cout * sizeof(float), stream);
  hipMemsetAsync(CNT, 0, (size_t)n * sizeof(float), stream);
  gmm_edge_kernel<<<(E + 7) / 8, 256, 0, stream>>>(Z, pseudo, ei, E, cout, mu, sigma, AGG, CNT);
  const int tot = n * cout;
  combine_kernel<<<(tot + 255) / 256, 256, 0, stream>>>(AGG, CNT, ZR, bias, out, n, cout, doRelu);
}
} // namespace

extern "C" void kernel_launch(void* const* d_in, const int* in_sizes, int n_in,
                              void* d_out, int out_size, void* d_ws, size_t ws_size,
                              hipStream_t stream) {
  (void)in_sizes; (void)n_in; (void)out_size; (void)ws_size;

  const float* x   = (const float*)d_in[0];
  const int*   ei6 = (const int*)d_in[1];  const float* ps6 = (const float*)d_in[2];
  const int*   ei5 = (const int*)d_in[3];  const float* ps5 = (const float*)d_in[4];
  const int*   ei4 = (const int*)d_in[5];  const float* ps4 = (const float*)d_in[6];
  const int*   ei3 = (const int*)d_in[7];  const float* ps3 = (const float*)d_in[8];
  const int*   ei2 = (const int*)d_in[9];  const float* ps2 = (const float*)d_in[10];
  const int*   hx6 = (const int*)d_in[11];
  const int*   hx5 = (const int*)d_in[12];
  const int*   hx4 = (const int*)d_in[13];
  const int*   hx3 = (const int*)d_in[14];
  // per-layer params: g, mu, sigma, root, bias at 15 + 5*(i-1)
  const float* G[9]; const float* MU[9]; const float* SG[9]; const float* RT[9]; const float* BI[9];
  for (int i = 0; i < 9; ++i) {
    G[i]  = (const float*)d_in[15 + 5 * i + 0];
    MU[i] = (const float*)d_in[15 + 5 * i + 1];
    SG[i] = (const float*)d_in[15 + 5 * i + 2];
    RT[i] = (const float*)d_in[15 + 5 * i + 3];
    BI[i] = (const float*)d_in[15 + 5 * i + 4];
  }

  // ---- workspace layout (fp32 words) ----
  float* ws = (float*)d_ws;
  size_t off = 0;
  float* X0 = ws + off; off += (size_t)N6 * 64;     // encoder skip features
  float* X1 = ws + off; off += (size_t)N5 * 64;
  float* X2 = ws + off; off += (size_t)N4 * 128;
  float* X3 = ws + off; off += (size_t)N3 * 256;
  float* X4 = ws + off; off += (size_t)N2 * 512;
  int* I1 = (int*)(ws + off); off += (size_t)N5 * 64;   // pool argmax indices
  int* I2 = (int*)(ws + off); off += (size_t)N4 * 128;
  int* I3 = (int*)(ws + off); off += (size_t)N3 * 256;
  int* I4 = (int*)(ws + off); off += (size_t)N2 * 512;
  float* Z   = ws + off; off += (size_t)N6 * 192;   // max 3*cout GEMM output
  float* ZR  = ws + off; off += (size_t)N6 * 64;    // max root GEMM output
  float* AGG = ws + off; off += (size_t)N6 * 64;    // max edge aggregation
  float* CNT = ws + off; off += (size_t)N6;
  float* CIN = ws + off; off += (size_t)N6 * 68;    // max decoder concat input
  float* H   = ws + off; off += (size_t)N5 * 128;   // max transient conv output

  // ---- encoder ----
  run_gmm_conv(x,  N6,   4,  64, ei6, E6, ps6, G[0], MU[0], SG[0], RT[0], BI[0], X0, 1, Z, ZR, AGG, CNT, stream);
  hex_pool_kernel<<<(N5 * 64 + 255) / 256, 256, 0, stream>>>(X0, hx6, 64, N5, X1, I1);
  run_gmm_conv(X1, N5,  64, 128, ei5, E5, ps5, G[1], MU[1], SG[1], RT[1], BI[1], H, 1, Z, ZR, AGG, CNT, stream);
  hex_pool_kernel<<<(N4 * 128 + 255) / 256, 256, 0, stream>>>(H, hx5, 128, N4, X2, I2);
  run_gmm_conv(X2, N4, 128, 256, ei4, E4, ps4, G[2], MU[2], SG[2], RT[2], BI[2], H, 1, Z, ZR, AGG, CNT, stream);
  hex_pool_kernel<<<(N3 * 256 + 255) / 256, 256, 0, stream>>>(H, hx4, 256, N3, X3, I3);
  run_gmm_conv(X3, N3, 256, 512, ei3, E3, ps3, G[3], MU[3], SG[3], RT[3], BI[3], H, 1, Z, ZR, AGG, CNT, stream);
  hex_pool_kernel<<<(N2 * 512 + 255) / 256, 256, 0, stream>>>(H, hx3, 512, N2, X4, I4);

  // ---- bottleneck ----
  run_gmm_conv(X4, N2, 512, 512, ei2, E2, ps2, G[4], MU[4], SG[4], RT[4], BI[4], H, 1, Z, ZR, AGG, CNT, stream);

  // ---- decoder: unpool + concat skip, then conv ----
  unpool_prep_kernel<<<(N3 * 768 + 255) / 256, 256, 0, stream>>>(X3, CIN, N3, 512, 768);
  unpool_scatter_kernel<<<(N2 * 512 + 255) / 256, 256, 0, stream>>>(H, I4, CIN, N2, 512, 768);
  run_gmm_conv(CIN, N3, 768, 256, ei3, E3, ps3, G[5], MU[5], SG[5], RT[5], BI[5], H, 1, Z, ZR, AGG, CNT, stream);

  unpool_prep_kernel<<<(N4 * 384 + 255) / 256, 256, 0, stream>>>(X2, CIN, N4, 256, 384);
  unpool_scatter_kernel<<<(N3 * 256 + 255) / 256, 256, 0, stream>>>(H, I3, CIN, N3, 256, 384);
  run_gmm_conv(CIN, N4, 384, 128, ei4, E4, ps4, G[6], MU[6], SG[6], RT[6], BI[6], H, 1, Z, ZR, AGG, CNT, stream);

  unpool_prep_kernel<<<(N5 * 192 + 255) / 256, 256, 0, stream>>>(X1, CIN, N5, 128, 192);
  unpool_scatter_kernel<<<(N4 * 128 + 255) / 256, 256, 0, stream>>>(H, I2, CIN, N4, 128, 192);
  run_gmm_conv(CIN, N5, 192, 64, ei5, E5, ps5, G[7], MU[7], SG[7], RT[7], BI[7], H, 1, Z, ZR, AGG, CNT, stream);

  // final skip must be 4 channels for g9 (cin = F0 + IN_CH = 68): use raw input x
  unpool_prep_kernel<<<(N6 * 68 + 255) / 256, 256, 0, stream>>>(x, CIN, N6, 64, 68);
  unpool_scatter_kernel<<<(N5 * 64 + 255) / 256, 256, 0, stream>>>(H, I1, CIN, N5, 64, 68);
  run_gmm_conv(CIN, N6, 68, 21, ei6, E6, ps6, G[8], MU[8], SG[8], RT[8], BI[8], H, 0, Z, ZR, AGG, CNT, stream);

  softmax21_kernel<<<(N6 + 255) / 256, 256, 0, stream>>>(H, (float*)d_out, N6);
}